// Attention_16793322127576
// MI455X (gfx1250) — compile-verified
//
#include <hip/hip_runtime.h>

typedef __attribute__((ext_vector_type(2))) float v2f;
typedef __attribute__((ext_vector_type(4))) float v4f;
typedef __attribute__((ext_vector_type(8))) float v8f;

#define B_    32
#define H_    32
#define KH_   8
#define D_    128
#define BS_   256
#define MB_   16
#define REP_  4
#define NWAVE 4
#define SCALE_ 0.08838834764831845f

__launch_bounds__(128, 1)
__global__ void paged_attn_decode(
    const float* __restrict__ q,
    const float* __restrict__ knew,
    const float* __restrict__ vnew,
    const float* __restrict__ k_cache,
    const float* __restrict__ v_cache,
    const int*   __restrict__ block_tables,
    const int*   __restrict__ context_lens,
    float*       __restrict__ out)
{
  __shared__ float s_q[REP_ * D_];
  __shared__ int   s_bt[MB_];
  __shared__ float s_p[NWAVE][16 * 16];
  __shared__ float s_v[NWAVE][16 * D_];
  __shared__ float s_acc[NWAVE][REP_][D_];
  __shared__ float s_m[NWAVE][REP_];
  __shared__ float s_l[NWAVE][REP_];

  const int tid  = threadIdx.x;
  const int lane = tid & 31;
  const int wave = tid >> 5;
  const int col  = lane & 15;   // token column (B/C) or M row (A)
  const int kh   = lane >> 4;   // which K-pair half of the fragment

  const int bg = blockIdx.x;
  const int b  = bg / KH_;
  const int g  = bg % KH_;
  const int L  = context_lens[b];

  // Stage the 4 query rows for this kv-head group (contiguous 512 floats).
  for (int i = tid; i < REP_ * D_; i += blockDim.x)
    s_q[i] = q[(size_t)b * (H_ * D_) + g * REP_ * D_ + i];
  if (tid < MB_) s_bt[tid] = block_tables[b * MB_ + tid];
  // Zero P tile once; rows 0..3 are rewritten each tile, rows 4..15 stay 0.
  for (int i = lane; i < 256; i += 32) s_p[wave][i] = 0.0f;
  __syncthreads();

  float* p_w = s_p[wave];
  float* v_w = s_v[wave];
  // LDS byte offset of this wave's V tile, plus this lane's 16B chunk.
  const unsigned v_lds_base =
      (unsigned)(unsigned long long)(const void*)v_w + (unsigned)(lane * 16);

  float m[REP_], l[REP_];
#pragma unroll
  for (int r = 0; r < REP_; ++r) { m[r] = -__builtin_inff(); l[r] = 0.0f; }
  v8f acc[8] = {};  // PV accumulators, one 16x16 C tile per 16-wide D slice

  const float* kq_new = knew + (size_t)(b * KH_ + g) * D_;
  const float* vq_new = vnew + (size_t)(b * KH_ + g) * D_;

  for (int p0 = wave * 16; p0 < L; p0 += NWAVE * 16) {
    // ---- Kick off V tile staging FIRST via CDNA5 async global->LDS copies.
    // One instruction moves a full 512B token row (32 lanes x 16B); all 16
    // rows are in flight at once on ASYNCcnt (no VGPR round-trip, no
    // per-row s_wait_loadcnt serialization).
    asm volatile("s_wait_dscnt 0" ::: "memory");  // prior-tile v_w reads done
#pragma unroll
    for (int j = 0; j < 16; ++j) {
      int pj = p0 + j;
      int blk = s_bt[pj >> 8];
      const float* vrow =
          v_cache + ((size_t)(blk * BS_ + (pj & (BS_ - 1))) * KH_ + g) * D_;
      if (pj == L - 1) vrow = vq_new;   // freshly decoded token substitution
      unsigned long long gaddr =
          (unsigned long long)vrow + (unsigned long long)(lane * 16);
      unsigned ldsa = v_lds_base + (unsigned)(j * (D_ * 4));
      asm volatile("global_load_async_to_lds_b128 %0, %1, off"
                   :: "v"(ldsa), "v"(gaddr) : "memory");
    }

    // ---- Per-lane K row pointer (token = col, shared by lane and lane+16).
    const int p = p0 + col;
    const float* krow;
    {
      int blk = s_bt[p >> 8];
      krow = k_cache + ((size_t)(blk * BS_ + (p & (BS_ - 1))) * KH_ + g) * D_;
      if (p == L - 1) krow = kq_new;
    }

    // ---- QK^T via V_WMMA_F32_16X16X4_F32: A = q (rows 0..3), B = K^T tile.
    // K is stream-once -> non-temporal loads keep L2 clean.
    v8f cqk = {};
#pragma unroll
    for (int d0 = 0; d0 < D_; d0 += 4) {
      v2f a = *(const v2f*)(s_q + (col & (REP_ - 1)) * D_ + d0 + kh * 2);
      if (col >= REP_) { a.x = 0.0f; a.y = 0.0f; }
      v2f bk = __builtin_nontemporal_load((const v2f*)(krow + d0 + kh * 2));
      cqk = __builtin_amdgcn_wmma_f32_16x16x4_f32(false, a, false, bk,
                                                  (short)0, cqk, false, false);
    }

    // ---- Online softmax per query row (scores: C elem r, lanes 0..15).
    const bool valid = (p0 + col) < L;
    float alpha[REP_];
#pragma unroll
    for (int r = 0; r < REP_; ++r) {
      float sc = cqk[r] * SCALE_;
      float s  = valid ? sc : -__builtin_inff();
      float mx = s;
#pragma unroll
      for (int off = 8; off >= 1; off >>= 1)
        mx = fmaxf(mx, __shfl_xor(mx, off, 16));
      float mn = fmaxf(m[r], mx);            // finite: tile has >=1 valid token
      alpha[r] = __expf(m[r] - mn);          // first tile: exp(-inf)=0
      float pv = valid ? __expf(sc - mn) : 0.0f;
      float ps = pv;
#pragma unroll
      for (int off = 8; off >= 1; off >>= 1)
        ps += __shfl_xor(ps, off, 16);
      l[r] = l[r] * alpha[r] + ps;
      m[r] = mn;
      if (lane < 16) p_w[r * 16 + lane] = pv;   // C-layout -> LDS
    }
#pragma unroll
    for (int n = 0; n < 8; ++n)
#pragma unroll
      for (int r = 0; r < REP_; ++r)
        acc[n][r] *= alpha[r];

    // ---- Fence the async V copies, then PV fragment reads from LDS.
#if __has_builtin(__builtin_amdgcn_s_wait_asynccnt)
    __builtin_amdgcn_s_wait_asynccnt(0);
#else
    asm volatile("s_wait_asynccnt 0" ::: "memory");
#endif

    // ---- PV: acc[n] += P(16x4 chunks) x V(4 x 16-wide D slice).
    v2f ap[4];
#pragma unroll
    for (int t = 0; t < 4; ++t)
      ap[t] = *(const v2f*)(p_w + col * 16 + t * 4 + kh * 2);   // A-layout
#pragma unroll
    for (int n = 0; n < 8; ++n) {
#pragma unroll
      for (int t = 0; t < 4; ++t) {
        v2f bv;
        bv.x = v_w[(t * 4 + kh * 2    ) * D_ + n * 16 + col];
        bv.y = v_w[(t * 4 + kh * 2 + 1) * D_ + n * 16 + col];
        acc[n] = __builtin_amdgcn_wmma_f32_16x16x4_f32(false, ap[t], false, bv,
                                                       (short)0, acc[n],
                                                       false, false);
      }
    }
  }

  // ---- Per-wave flash partials -> LDS.
#pragma unroll
  for (int n = 0; n < 8; ++n)
#pragma unroll
    for (int r = 0; r < REP_; ++r)
      if (lane < 16) s_acc[wave][r][n * 16 + lane] = acc[n][r];
  if (lane == 0) {
#pragma unroll
    for (int r = 0; r < REP_; ++r) { s_m[wave][r] = m[r]; s_l[wave][r] = l[r]; }
  }
  __syncthreads();

  // ---- Merge 4 wave partials and write output (h = g*REP + r).
  for (int i = tid; i < REP_ * D_; i += blockDim.x) {
    int r = i >> 7, d = i & (D_ - 1);
    float M = -__builtin_inff();
#pragma unroll
    for (int w = 0; w < NWAVE; ++w) M = fmaxf(M, s_m[w][r]);
    float Ls = 0.0f, val = 0.0f;
#pragma unroll
    for (int w = 0; w < NWAVE; ++w) {
      float wg = __expf(s_m[w][r] - M);   // empty waves: exp(-inf)=0
      Ls  += s_l[w][r] * wg;
      val += s_acc[w][r][d] * wg;
    }
    out[(size_t)b * (H_ * D_) + (g * REP_ + r) * D_ + d] = val / Ls;
  }
}

extern "C" void kernel_launch(void* const* d_in, const int* in_sizes, int n_in,
                              void* d_out, int out_size, void* d_ws, size_t ws_size,
                              hipStream_t stream) {
  (void)in_sizes; (void)n_in; (void)d_ws; (void)ws_size; (void)out_size;
  const float* q        = (const float*)d_in[0];
  const float* k        = (const float*)d_in[1];
  const float* v        = (const float*)d_in[2];
  const float* k_cache  = (const float*)d_in[3];
  const float* v_cache  = (const float*)d_in[4];
  const int*   btab     = (const int*)d_in[5];
  const int*   ctx_lens = (const int*)d_in[6];
  // d_in[7] (slot_mapping) not needed: substituting the new k/v at position
  // L-1 with disjoint per-sequence block tables equals the reference scatter.
  float* out = (float*)d_out;

  paged_attn_decode<<<dim3(B_ * KH_), dim3(NWAVE * 32), 0, stream>>>(
      q, k, v, k_cache, v_cache, btab, ctx_lens, out);
}